// GAT_81398220194149
// MI455X (gfx1250) — compile-verified
//
#include <hip/hip_runtime.h>
#include <hip/hip_bf16.h>

// ---------------------------------------------------------------------------
// GAT (3x GATConv, N=100000, D=128, H=4, F=32) for gfx1250 (MI455X, wave32).
// GEMMs run on V_WMMA_F32_16X16X4_F32 (full fp32, matrix pipe); edge phase is
// HBM-bound gather/scatter with native f32 global atomics.
// ---------------------------------------------------------------------------

#define DLAT 128
#define HEADS 4
#define FEATS 32
#define LDS_STRIDE 132   // 16-row A tile, padded to kill 64-bank conflicts

typedef __attribute__((ext_vector_type(2))) float v2f;
typedef __attribute__((ext_vector_type(8))) float v8f;

__device__ __forceinline__ float leaky(float x, float slope) {
    return x >= 0.0f ? x : slope * x;
}

__device__ __forceinline__ void atomicMaxF32(float* addr, float val) {
    unsigned int* ua = (unsigned int*)addr;
    unsigned int old = *ua;
    while (__uint_as_float(old) < val) {
        unsigned int assumed = old;
        old = atomicCAS(ua, assumed, __float_as_uint(val));
        if (old == assumed) break;
    }
}

// x[n,d] = weights[n] * lin_W[d] + lin_b[d]
__global__ void gat_init_x(const float* __restrict__ w,
                           const float* __restrict__ linW,
                           const float* __restrict__ linB,
                           float* __restrict__ x, int nd) {
    int t = blockIdx.x * blockDim.x + threadIdx.x;
    if (t >= nd) return;
    int n = t >> 7, d = t & (DLAT - 1);
    x[t] = w[n] * linW[d] + linB[d];
}

// h = (leaky?) x @ W^T   — one block = 16 rows, wave w = output cols 16w..16w+15
__global__ void __launch_bounds__(256)
gat_gemm_wmma(const float* __restrict__ X, const float* __restrict__ W,
              float* __restrict__ Hout, int N, int leakyIn) {
    __shared__ float tileA[16 * LDS_STRIDE];
    const int tid  = threadIdx.x;
    const int row0 = blockIdx.x * 16;

    // cooperative stage of the 16x128 A tile (8 floats/thread)
    for (int i = 0; i < 8; ++i) {
        int idx = tid + i * 256;              // 0..2047
        int r = idx >> 7, c = idx & (DLAT - 1);
        int gr = row0 + r;
        float v = (gr < N) ? X[(size_t)gr * DLAT + c] : 0.0f;
        if (leakyIn) v = leaky(v, 0.01f);
        tileA[r * LDS_STRIDE + c] = v;
    }
    __syncthreads();

    const int wave  = tid >> 5;               // 0..7 -> column tile
    const int lane  = tid & 31;
    const int mrow  = lane & 15;              // A row within tile / also D col
    const int khalf = (lane >> 4) << 1;       // 0 or 2 (ISA 16x4 f32 A layout)
    const int coln  = 16 * wave + (lane & 15);
    const float* wrow = W + (size_t)coln * DLAT;   // fc_W[coln][*] (B = W^T)

    v8f acc = {};
    #pragma unroll
    for (int kk = 0; kk < DLAT; kk += 4) {
        const int k = kk + khalf;
        v2f a, b;
        a[0] = tileA[mrow * LDS_STRIDE + k];
        a[1] = tileA[mrow * LDS_STRIDE + k + 1];
        b[0] = wrow[k];
        b[1] = wrow[k + 1];
        acc = __builtin_amdgcn_wmma_f32_16x16x4_f32(
            /*neg_a=*/false, a, /*neg_b=*/false, b,
            /*c_mod=*/(short)0, acc, /*reuse_a=*/false, /*reuse_b=*/false);
    }

    // D layout: VGPR v -> M=v (lanes 0-15) / M=v+8 (lanes 16-31), N=lane%16
    const int rbase = row0 + ((lane >> 4) << 3);
    #pragma unroll
    for (int v = 0; v < 8; ++v) {
        int gr = rbase + v;
        if (gr < N) Hout[(size_t)gr * DLAT + coln] = acc[v];
    }
}

// el/er: one wave per node; 8-lane head-group shuffle reduction (wave32)
__global__ void __launch_bounds__(256)
gat_attn_scores(const float* __restrict__ Hmat,
                const float* __restrict__ al, const float* __restrict__ ar,
                float* __restrict__ el, float* __restrict__ er, int N) {
    int t = blockIdx.x * blockDim.x + threadIdx.x;
    int node = t >> 5, lane = t & 31;
    if (node >= N) return;                     // wave-uniform
    const float4 h4 = *(const float4*)(Hmat + (size_t)node * DLAT + 4 * lane);
    const float4 a4 = *(const float4*)(al + 4 * lane);
    const float4 b4 = *(const float4*)(ar + 4 * lane);
    float pl = h4.x * a4.x + h4.y * a4.y + h4.z * a4.z + h4.w * a4.w;
    float pr = h4.x * b4.x + h4.y * b4.y + h4.z * b4.z + h4.w * b4.w;
    #pragma unroll
    for (int m = 1; m < 8; m <<= 1) {
        pl += __shfl_xor(pl, m, 32);
        pr += __shfl_xor(pr, m, 32);
    }
    if ((lane & 7) == 0) {
        int head = lane >> 3;
        el[(size_t)node * HEADS + head] = pl;
        er[(size_t)node * HEADS + head] = pr;
    }
}

// out <- bias broadcast; m <- -inf; s <- 0
__global__ void gat_init_layer(float* __restrict__ out, const float* __restrict__ bias,
                               float* __restrict__ m, float* __restrict__ s,
                               int nd, int nh) {
    int t = blockIdx.x * blockDim.x + threadIdx.x;
    if (t < nd) out[t] = bias[t & (DLAT - 1)];
    if (t < nh) { m[t] = -3.402823466e38f; s[t] = 0.0f; }
}

// segment max of leaky(el[src]+er[dst]) over dst; edge e>=E is self loop e-E
__global__ void gat_edge_max(const float* __restrict__ el, const float* __restrict__ er,
                             const int* __restrict__ src, const int* __restrict__ dst,
                             float* __restrict__ m, int E, int N) {
    int t = blockIdx.x * blockDim.x + threadIdx.x;
    int total = (E + N) * HEADS;
    if (t >= total) return;
    int e = t >> 2, hh = t & 3;
    int s_, d_;
    if (e < E) { s_ = src[e]; d_ = dst[e]; } else { s_ = d_ = e - E; }
    float v = leaky(el[(size_t)s_ * HEADS + hh] + er[(size_t)d_ * HEADS + hh], 0.2f);
    atomicMaxF32(&m[(size_t)d_ * HEADS + hh], v);
}

// a = exp(e - m[dst]); cache a; segment-sum into s
__global__ void gat_edge_sum(const float* __restrict__ el, const float* __restrict__ er,
                             const int* __restrict__ src, const int* __restrict__ dst,
                             const float* __restrict__ m, float* __restrict__ s,
                             float* __restrict__ abuf, int E, int N) {
    int t = blockIdx.x * blockDim.x + threadIdx.x;
    int total = (E + N) * HEADS;
    if (t >= total) return;
    int e = t >> 2, hh = t & 3;
    int s_, d_;
    if (e < E) { s_ = src[e]; d_ = dst[e]; } else { s_ = d_ = e - E; }
    float v = leaky(el[(size_t)s_ * HEADS + hh] + er[(size_t)d_ * HEADS + hh], 0.2f);
    float a = __expf(v - m[(size_t)d_ * HEADS + hh]);
    abuf[t] = a;
    atomicAdd(&s[(size_t)d_ * HEADS + hh], a);
}

// out[dst] += h[src] * (a/s[dst]) — one wave per edge, 32 lanes x float4 = 128 wide
__global__ void __launch_bounds__(256)
gat_aggregate(const float* __restrict__ Hmat, const float* __restrict__ abuf,
              const float* __restrict__ ssum,
              const int* __restrict__ src, const int* __restrict__ dst,
              float* __restrict__ out, int E, int N) {
    int t = blockIdx.x * blockDim.x + threadIdx.x;
    int e = t >> 5, lane = t & 31;
    if (e >= E + N) return;                    // wave-uniform
    int s_, d_;
    if (e < E) { s_ = src[e]; d_ = dst[e]; } else { s_ = d_ = e - E; }
    int head = lane >> 3;
    float alpha = abuf[(size_t)e * HEADS + head] /
                  ssum[(size_t)d_ * HEADS + head];
    const float4 h4 = *(const float4*)(Hmat + (size_t)s_ * DLAT + 4 * lane);
    float* o = out + (size_t)d_ * DLAT + 4 * lane;
    atomicAdd(o + 0, h4.x * alpha);
    atomicAdd(o + 1, h4.y * alpha);
    atomicAdd(o + 2, h4.z * alpha);
    atomicAdd(o + 3, h4.w * alpha);
}

// logits[n] = x[n,:] . pred_W + pred_b — one wave per node
__global__ void __launch_bounds__(256)
gat_pred(const float* __restrict__ X, const float* __restrict__ pW,
         const float* __restrict__ pb, float* __restrict__ outp, int N) {
    int t = blockIdx.x * blockDim.x + threadIdx.x;
    int node = t >> 5, lane = t & 31;
    if (node >= N) return;                     // wave-uniform
    const float4 x4 = *(const float4*)(X + (size_t)node * DLAT + 4 * lane);
    const float4 w4 = *(const float4*)(pW + 4 * lane);
    float p = x4.x * w4.x + x4.y * w4.y + x4.z * w4.z + x4.w * w4.w;
    #pragma unroll
    for (int m = 1; m < 32; m <<= 1) p += __shfl_xor(p, m, 32);
    if (lane == 0) outp[node] = p + pb[0];
}

static inline int cdiv_i(long long a, long long b) { return (int)((a + b - 1) / b); }

extern "C" void kernel_launch(void* const* d_in, const int* in_sizes, int n_in,
                              void* d_out, int out_size, void* d_ws, size_t ws_size,
                              hipStream_t stream) {
    const float* weights = (const float*)d_in[0];
    const float* lin_W   = (const float*)d_in[1];
    const float* lin_b   = (const float*)d_in[2];
    const float* fc_W    = (const float*)d_in[3];
    const float* attn_l  = (const float*)d_in[4];
    const float* attn_r  = (const float*)d_in[5];
    const float* conv_b  = (const float*)d_in[6];
    const float* pred_W  = (const float*)d_in[7];
    const float* pred_b  = (const float*)d_in[8];
    const int*   src     = (const int*)d_in[9];
    const int*   dst     = (const int*)d_in[10];

    const int N = in_sizes[0];
    const int E = in_sizes[9];
    const int Etot = E + N;
    const int nd = N * DLAT;
    const int nh = N * HEADS;

    // workspace layout (floats)
    float* ws   = (float*)d_ws;
    float* bufA = ws;                       // x ping
    float* bufB = bufA + (size_t)nd;        // h (per-layer features)
    float* bufC = bufB + (size_t)nd;        // x pong / out accumulator
    float* el   = bufC + (size_t)nd;
    float* er   = el + (size_t)nh;
    float* mseg = er + (size_t)nh;
    float* sseg = mseg + (size_t)nh;
    float* abuf = sseg + (size_t)nh;        // [Etot, H] softmax numerators

    float* xcur = bufA;
    float* hbuf = bufB;
    float* xout = bufC;

    gat_init_x<<<cdiv_i(nd, 256), 256, 0, stream>>>(weights, lin_W, lin_b, xcur, nd);

    for (int layer = 0; layer < 3; ++layer) {
        const float* W  = fc_W   + (size_t)layer * DLAT * DLAT;
        const float* al = attn_l + (size_t)layer * HEADS * FEATS;
        const float* ar = attn_r + (size_t)layer * HEADS * FEATS;
        const float* bi = conv_b + (size_t)layer * DLAT;

        gat_gemm_wmma<<<cdiv_i(N, 16), 256, 0, stream>>>(xcur, W, hbuf, N, layer > 0);
        gat_attn_scores<<<cdiv_i((long long)N * 32, 256), 256, 0, stream>>>(
            hbuf, al, ar, el, er, N);
        gat_init_layer<<<cdiv_i(nd, 256), 256, 0, stream>>>(xout, bi, mseg, sseg, nd, nh);
        gat_edge_max<<<cdiv_i((long long)Etot * HEADS, 256), 256, 0, stream>>>(
            el, er, src, dst, mseg, E, N);
        gat_edge_sum<<<cdiv_i((long long)Etot * HEADS, 256), 256, 0, stream>>>(
            el, er, src, dst, mseg, sseg, abuf, E, N);
        gat_aggregate<<<cdiv_i((long long)Etot * 32, 256), 256, 0, stream>>>(
            hbuf, abuf, sseg, src, dst, xout, E, N);

        float* tmp = xcur; xcur = xout; xout = tmp;   // ping-pong
    }

    gat_pred<<<cdiv_i((long long)N * 32, 256), 256, 0, stream>>>(
        xcur, pred_W, pred_b, (float*)d_out, N);
}